// ContinuousConv1D_33492154974874
// MI455X (gfx1250) — compile-verified
//
#include <hip/hip_runtime.h>

// Problem constants (from reference)
#define CCH 16          // channels
#define EEV 65536       // events per channel
#define KS  5           // kernel size
#define OC  64          // out channels
#define RR  98          // ref size
#define BB  256         // batch
#define NROWS (RR*BB)   // 25088 rows in scatter target
#define KC  (KS*CCH)    // 80 = reduction dim of the GEMM
#define S_ELEMS ((size_t)NROWS*KC)  // 2,007,040 f32 = ~8 MB scratch

#define NPAIR (KC/2)    // 40 k-pairs
#define LDST  80        // float2 row stride in LDS: 160 dwords == 32 mod 64 banks
                        // -> half-waves hit banks 0..31 / 32..63, conflict-free

typedef float v2f __attribute__((ext_vector_type(2)));
typedef float v8f __attribute__((ext_vector_type(8)));

static_assert(NROWS % 16 == 0, "M tiles exact");
static_assert((NROWS/16) % 8 == 0, "8 waves per block exact");
static_assert((CCH*EEV) % (256*4) == 0, "phase-1 vectorization exact");

// ---------------- Phase 0: zero the coefficient accumulator ----------------
__global__ void cc_zero_kernel(float4* __restrict__ S4) {
    int i = blockIdx.x * blockDim.x + threadIdx.x;
    S4[i] = make_float4(0.f, 0.f, 0.f, 0.f);
}

// ---------------- Phase 1: scatter spline coefficients ---------------------
// 4 events per thread (b128 loads). Each event: two scalar f32 atomics into
// S[r, k*16+c]. E = 65536 so channel = event_index >> 16.
__global__ void __launch_bounds__(256)
cc_scatter_kernel(const float4* __restrict__ pseudo4,
                  const float4* __restrict__ y4,
                  const int4*   __restrict__ ref4,
                  float*        __restrict__ S) {
    int q  = blockIdx.x * blockDim.x + threadIdx.x;    // quad index
    int t0 = q * 4;                                    // first event index
    float4 pv = pseudo4[q];
    float4 yv = y4[q];
    int4   rv = ref4[q];

    const float p[4] = {pv.x, pv.y, pv.z, pv.w};
    const float yy[4] = {yv.x, yv.y, yv.z, yv.w};
    const int   rr[4] = {rv.x, rv.y, rv.z, rv.w};

    #pragma unroll
    for (int j = 0; j < 4; ++j) {
        int   c    = (t0 + j) >> 16;                   // channel
        float v    = p[j] * (float)(KS - 1);
        float bot  = floorf(v);
        float frac = v - bot;
        int   i0   = (int)bot;
        int   i1   = i0 + 1; if (i1 >= KS) i1 = 0;     // matches (i0+1)%K
        float* row = S + (size_t)rr[j] * KC;
        atomicAdd(row + i0 * CCH + c, yy[j] * (1.0f - frac));
        atomicAdd(row + i1 * CCH + c, yy[j] * frac);
    }
}

// ---------------- Phase 2: WMMA GEMM  conv = S[25088x80] @ W[80x64] --------
// One wave per 16x64 output tile (4 f32 accumulators), 8 waves/block.
// W staged in LDS pre-swizzled as float2 k-pairs so each B fragment is a
// single aligned ds_load_b64 (no VGPR re-assembly moves).
__global__ void __launch_bounds__(256)
cc_gemm_kernel(const float* __restrict__ S,
               const float* __restrict__ weight,   // [80*64] row-major
               const float* __restrict__ bias,     // [64]
               float*       __restrict__ out) {    // [B,O,R]
    __shared__ float2 Wswz[NPAIR * LDST];          // 25.6 KB
    __shared__ float  biasS[OC];

    // stage W swizzled: Wswz[p*LDST + col] = { W[2p][col], W[2p+1][col] }
    for (int i = threadIdx.x; i < NPAIR * OC; i += 256) {
        int p = i >> 6, col = i & 63;
        Wswz[p * LDST + col] =
            make_float2(weight[(2*p)     * OC + col],
                        weight[(2*p + 1) * OC + col]);
    }
    if (threadIdx.x < OC) biasS[threadIdx.x] = bias[threadIdx.x];
    __syncthreads();

    const int lane = threadIdx.x & 31;
    const int wave = threadIdx.x >> 5;
    const int half = lane >> 4;        // 0: K pair {k0,k0+1}; 1: {k0+2,k0+3}
    const int l16  = lane & 15;
    const int m0   = (blockIdx.x * 8 + wave) * 16;   // output row tile base

    v8f acc0 = {}, acc1 = {}, acc2 = {}, acc3 = {};

    // A fragment source: lane<16 -> S[m0+l16, k0..k0+1]; lane>=16 -> k0+2..k0+3
    const float2* __restrict__ Arow =
        reinterpret_cast<const float2*>(S + (size_t)(m0 + l16) * KC) + half;

    #pragma unroll
    for (int k0 = 0; k0 < KC; k0 += 4) {
        float2 av = Arow[k0 >> 1];                 // 8B-aligned global_load_b64
        v2f a; a.x = av.x; a.y = av.y;

        const float2* brow = Wswz + ((k0 >> 1) + half) * LDST + l16;
        #pragma unroll
        for (int t = 0; t < 4; ++t) {
            float2 bv = brow[t * 16];              // single ds_load_b64
            v2f b; b.x = bv.x; b.y = bv.y;
            v8f* accp = (t == 0) ? &acc0 : (t == 1) ? &acc1 : (t == 2) ? &acc2 : &acc3;
            *accp = __builtin_amdgcn_wmma_f32_16x16x4_f32(
                        /*neg_a=*/false, a, /*neg_b=*/false, b,
                        /*c_mod=*/(short)0, *accp,
                        /*reuse_a=*/false, /*reuse_b=*/false);
        }
    }

    // Store with [B,O,R] transpose + bias. C/D layout: VGPR i, lanes 0-15: M=i,
    // lanes 16-31: M=i+8; N = lane&15 within the 16-wide n-tile.
    #pragma unroll
    for (int t = 0; t < 4; ++t) {
        const v8f& acc = (t == 0) ? acc0 : (t == 1) ? acc1 : (t == 2) ? acc2 : acc3;
        int n = t * 16 + l16;
        float bn = biasS[n];
        #pragma unroll
        for (int i = 0; i < 8; ++i) {
            int m  = m0 + i + half * 8;        // global conv row = b*R + r
            int bi = m / RR;
            int r  = m - bi * RR;
            out[(size_t)bi * (OC * RR) + (size_t)n * RR + r] = acc[i] + bn;
        }
    }
}

extern "C" void kernel_launch(void* const* d_in, const int* in_sizes, int n_in,
                              void* d_out, int out_size, void* d_ws, size_t ws_size,
                              hipStream_t stream) {
    const float* pseudo  = (const float*)d_in[0];   // [C,E,1]
    const float* y       = (const float*)d_in[1];   // [C,E,1]
    const float* weight  = (const float*)d_in[2];   // [K,C,O] == [80][64] flat
    const float* bias    = (const float*)d_in[3];   // [O]
    const int*   ref_idx = (const int*)d_in[4];     // [C,E]
    // d_in[5] = ref_deg (unused, norm=False), d_in[6] = batch_size (=256)

    float* S   = (float*)d_ws;                      // [25088][80] f32, 8 MB
    float* out = (float*)d_out;                     // [B,O,R]

    cc_zero_kernel   <<<(S_ELEMS/4)/256,   256, 0, stream>>>((float4*)S);
    cc_scatter_kernel<<<(CCH*EEV)/(256*4), 256, 0, stream>>>(
        (const float4*)pseudo, (const float4*)y, (const int4*)ref_idx, S);
    cc_gemm_kernel   <<<(NROWS/16)/8,      256, 0, stream>>>(S, weight, bias, out);
}